// funcRepresentation_47880295416555
// MI455X (gfx1250) — compile-verified
//
#include <hip/hip_runtime.h>

typedef __attribute__((ext_vector_type(2))) float v2f;
typedef __attribute__((ext_vector_type(8))) float v8f;

#define B_  32
#define M_  12
#define T_  512
#define G_  256
#define MP  16          // M padded to WMMA tile height
#define YS  516         // LDS float row stride (516 % 64 == 4 -> conflict-free)
#define MS  516         // LDS byte row stride for mask

__launch_bounds__(256, 1)
__global__ void func_repr_kernel(const float* __restrict__ y,
                                 const float* __restrict__ x,
                                 const unsigned char* __restrict__ mask,
                                 const float* __restrict__ grid,
                                 const float* __restrict__ sigma,
                                 float* __restrict__ out)
{
    __shared__ float         s_y[MP * YS];   // ~33 KB, rows 12..15 zero
    __shared__ unsigned char s_m[MP * MS];   // ~8 KB,  rows 12..15 zero
    __shared__ float         s_x[T_];        // 2 KB

    const int b    = blockIdx.x;
    const int tid  = threadIdx.x;
    const int lane = tid & 31;
    const int wave = tid >> 5;
    const int half = lane >> 4;   // 0: lanes 0-15, 1: lanes 16-31
    const int idx  = lane & 15;

    // ---- stage A-matrices (padded to 16 rows) and x into LDS ----
    for (int i = tid; i < MP * T_; i += 256) {
        const int m = i >> 9;          // / 512
        const int t = i & (T_ - 1);
        float yv = 0.0f;
        unsigned char mv = 0;
        if (m < M_) {
            const int gidx = (b * M_ + m) * T_ + t;
            yv = y[gidx];
            mv = mask[gidx];
        }
        s_y[m * YS + t] = yv;
        s_m[m * MS + t] = mv;
    }
    for (int i = tid; i < T_; i += 256) s_x[i] = x[b * T_ + i];

    const float s0 = sigma[0];
    const float s1 = sigma[1];
    const float LOG2E = 1.4426950408889634f;
    const float k0 = -0.5f * LOG2E / (s0 * s0);   // density channel (mask)
    const float k1 = -0.5f * LOG2E / (s1 * s1);   // conv channel (y)
    const bool same_sigma = (k0 == k1);           // uniform across the block

    __syncthreads();

    // ---- each wave computes two 16-wide g tiles, full K = 512 ----
    for (int i = 0; i < 2; ++i) {
        const int gt = wave * 2 + i;          // 8 waves * 2 = 16 tiles = G/16
        const int g  = gt * 16 + idx;
        const float dg = grid[g];

        v8f acc0 = {};   // density = mask x W0
        v8f acc1 = {};   // conv    = y    x W1

        const unsigned char* rowm = &s_m[idx * MS + 2 * half];
        const float*         rowy = &s_y[idx * YS + 2 * half];
        const float*         px   = &s_x[2 * half];

        if (same_sigma) {
            // Common case: both channels share W -> one B fragment, 2 exps per
            // K-step, two WMMAs off the same operand (halves TRANS-pipe work).
            for (int t0 = 0; t0 < T_; t0 += 4) {
                const float2 xx = *(const float2*)(px + t0);
                const float d0 = dg - xx.x;
                const float d1 = dg - xx.y;
                v2f bw;
                bw.x = __builtin_amdgcn_exp2f(d0 * d0 * k0);
                bw.y = __builtin_amdgcn_exp2f(d1 * d1 * k0);

                const uchar2 mm = *(const uchar2*)(rowm + t0);
                v2f a0, a1;
                a0.x = (float)mm.x;
                a0.y = (float)mm.y;
                a1 = *(const v2f*)(rowy + t0);

                acc0 = __builtin_amdgcn_wmma_f32_16x16x4_f32(
                           false, a0, false, bw, (short)0, acc0, false, false);
                acc1 = __builtin_amdgcn_wmma_f32_16x16x4_f32(
                           false, a1, false, bw, (short)0, acc1, false, false);
            }
        } else {
            // General case: per-channel Gaussian widths.
            for (int t0 = 0; t0 < T_; t0 += 4) {
                const float2 xx = *(const float2*)(px + t0);
                const float d0 = dg - xx.x;
                const float d1 = dg - xx.y;
                const float q0 = d0 * d0;
                const float q1 = d1 * d1;
                v2f b0, b1;
                b0.x = __builtin_amdgcn_exp2f(q0 * k0);
                b0.y = __builtin_amdgcn_exp2f(q1 * k0);
                b1.x = __builtin_amdgcn_exp2f(q0 * k1);
                b1.y = __builtin_amdgcn_exp2f(q1 * k1);

                const uchar2 mm = *(const uchar2*)(rowm + t0);
                v2f a0, a1;
                a0.x = (float)mm.x;
                a0.y = (float)mm.y;
                a1 = *(const v2f*)(rowy + t0);

                acc0 = __builtin_amdgcn_wmma_f32_16x16x4_f32(
                           false, a0, false, b0, (short)0, acc0, false, false);
                acc1 = __builtin_amdgcn_wmma_f32_16x16x4_f32(
                           false, a1, false, b1, (short)0, acc1, false, false);
            }
        }

        // ---- epilogue: D row r+8*half, col idx; drop padded rows >= 12 ----
        #pragma unroll
        for (int r = 0; r < 8; ++r) {
            const int mrow = r + 8 * half;
            if (mrow < M_) {
                const float dens  = acc0[r];
                const float ratio = acc1[r] * __builtin_amdgcn_rcpf(dens + 1e-8f);
                float2 v = make_float2(dens, ratio);
                *(float2*)&out[((size_t)(b * G_ + g)) * (2 * M_) + 2 * mrow] = v;
            }
        }
    }
}

extern "C" void kernel_launch(void* const* d_in, const int* in_sizes, int n_in,
                              void* d_out, int out_size, void* d_ws, size_t ws_size,
                              hipStream_t stream) {
    (void)in_sizes; (void)n_in; (void)out_size; (void)d_ws; (void)ws_size;
    const float*         y     = (const float*)d_in[0];
    const float*         x     = (const float*)d_in[1];
    const unsigned char* mask  = (const unsigned char*)d_in[2];
    const float*         grid  = (const float*)d_in[3];
    const float*         sigma = (const float*)d_in[4];
    float*               out   = (float*)d_out;

    func_repr_kernel<<<dim3(B_), dim3(256), 0, stream>>>(y, x, mask, grid, sigma, out);
}